// CausalSelfAttention_37056977829962
// MI455X (gfx1250) — compile-verified
//
#include <hip/hip_runtime.h>
#include <math.h>

// CDNA5 / gfx1250, wave32. FP32 WMMA (V_WMMA_F32_16X16X4_F32) everywhere.
typedef float v2f __attribute__((ext_vector_type(2)));
typedef float v8f __attribute__((ext_vector_type(8)));

constexpr int kB = 2;
constexpr int kN = 2048;
constexpr int kD = 1024;
constexpr int kH = 16;
constexpr int kHD = 64;     // head dim
constexpr int kE = 3 * kHD; // 192 (k,q,v per head)

// fp32 WMMA fragment layout (ISA 7.12.2):
//   A (16x4):  element (M = lane&15, K = v + 2*(lane>>4)), v in {0,1}  -> v2f
//   B (4x16):  element (K = v + 2*(lane>>4), N = lane&15), v in {0,1}  -> v2f
//   C/D 16x16: element (M = v + 8*(lane>>4), N = lane&15), v in 0..7   -> v8f
static __device__ __forceinline__ v8f wmma_f32(v2f a, v2f b, v8f c) {
  return __builtin_amdgcn_wmma_f32_16x16x4_f32(false, a, false, b, (short)0, c,
                                               false, false);
}

// --- async global->LDS staging (GLOBAL_LOAD_ASYNC_TO_LDS_B128, ASYNCcnt) ---
#if __has_builtin(__builtin_amdgcn_global_load_async_to_lds_b128) && \
    __has_builtin(__builtin_amdgcn_s_wait_asynccnt)
#define USE_ASYNC_LDS 1
typedef int v4i_vs __attribute__((__vector_size__(16)));  // 16-byte payload
typedef __attribute__((address_space(1))) v4i_vs as1_v4i;
typedef __attribute__((address_space(3))) v4i_vs as3_v4i;
#endif

// copy 16 bytes (one float4) global -> LDS
static __device__ __forceinline__ void lds_copy16(const float* g, float* l) {
#ifdef USE_ASYNC_LDS
  __builtin_amdgcn_global_load_async_to_lds_b128(
      (as1_v4i*)const_cast<float*>(g), (as3_v4i*)l, 0, 0);
#else
  *(float4*)l = *(const float4*)g;
#endif
}

static __device__ __forceinline__ void async_stage_wait() {
#ifdef USE_ASYNC_LDS
  __builtin_amdgcn_s_wait_asynccnt(0);
#endif
}

// ---------------------------------------------------------------------------
// Kernel 1: per-head kqv projection.
// kqv[b,h,n,e] = sum_d x[b,n,d] * Wkqv[h,e,d] + bkqv[h,e]
// Block: 128 threads (4 waves). Tile: 64 rows x 192 cols.
// Double-buffered async K-tile staging: loads of tile kt+64 overlap WMMAs on
// tile kt; s_wait_asynccnt 0 + barrier before buffer flip.
// ---------------------------------------------------------------------------
__global__ __launch_bounds__(128) void kqv_gemm_kernel(
    const float* __restrict__ x, const float* __restrict__ Wkqv,
    const float* __restrict__ bkqv, float* __restrict__ kqv) {
  constexpr int TS = 68;  // LDS row stride (68 % 64 == 4 -> conflict-free)
  __shared__ float xs[2][64 * TS];
  __shared__ float wsh[2][kE * TS];

  const int tid = threadIdx.x;
  const int w = tid >> 5;
  const int lane = tid & 31;
  const int half = lane >> 4;
  const int l16 = lane & 15;

  const int m0 = blockIdx.x * 64;
  const int h = blockIdx.y;
  const int b = blockIdx.z;

  const float* xbase = x + ((size_t)b * kN + m0) * kD;
  const float* wbase = Wkqv + (size_t)h * kE * kD;

  auto stage = [&](int bufi, int kt) {
    for (int i = tid; i < 64 * 16; i += 128) {
      int r = i >> 4, c = (i & 15) * 4;
      lds_copy16(&xbase[(size_t)r * kD + kt + c], &xs[bufi][r * TS + c]);
    }
    for (int i = tid; i < kE * 16; i += 128) {
      int r = i >> 4, c = (i & 15) * 4;
      lds_copy16(&wbase[(size_t)r * kD + kt + c], &wsh[bufi][r * TS + c]);
    }
  };

  v8f acc[12] = {};  // 12 e-tiles of 16 cols each (192 = 12*16)

  stage(0, 0);
  async_stage_wait();
  __syncthreads();

  int buf = 0;
  for (int kt = 0; kt < kD; kt += 64) {
    if (kt + 64 < kD) stage(buf ^ 1, kt + 64);  // overlap with compute below

#pragma unroll
    for (int kk = 0; kk < 64; kk += 4) {
      v2f a = *(const v2f*)&xs[buf][(w * 16 + l16) * TS + kk + 2 * half];
#pragma unroll
      for (int et = 0; et < 12; ++et) {
        v2f bf = *(const v2f*)&wsh[buf][(et * 16 + l16) * TS + kk + 2 * half];
        acc[et] = wmma_f32(a, bf, acc[et]);
      }
    }
    async_stage_wait();
    __syncthreads();
    buf ^= 1;
  }

  // bias + store kqv[b,h,row,e]
  const size_t rowbase = ((size_t)b * kH + h) * kN;
#pragma unroll
  for (int et = 0; et < 12; ++et) {
    int col = et * 16 + l16;
    float bias = bkqv[h * kE + col];
#pragma unroll
    for (int v = 0; v < 8; ++v) {
      int row = m0 + w * 16 + v + 8 * half;
      kqv[(rowbase + row) * kE + col] = acc[et][v] + bias;
    }
  }
}

// ---------------------------------------------------------------------------
// Kernel 2: causal attention for one 16-row strip of one (b,h).
// Full score strip (16 x 2048) lives in LDS (~131 KB of the 320 KB WGP LDS).
// 4 waves split column tiles (j % 4 == wave). Exact softmax, A written to
// global, O = A @ V accumulated with WMMA + cross-wave LDS reduction.
// ---------------------------------------------------------------------------
__global__ __launch_bounds__(128) void attn_kernel(
    const float* __restrict__ kqv, float* __restrict__ Aout,
    float* __restrict__ sa) {
  constexpr int SS = 2052;  // score row stride (2052 % 64 == 4 -> conflict-free)
  __shared__ float s[16 * SS];  // 131 KB
  __shared__ float red[16 * 8];
  __shared__ float rowmax[16];
  __shared__ float rowsum[16];
  __shared__ float ored[4 * 16 * 64];  // per-wave partial O

  const int tid = threadIdx.x;
  const int w = tid >> 5;
  const int lane = tid & 31;
  const int half = lane >> 4;
  const int l16 = lane & 15;

  const int it = blockIdx.x;  // row tile index
  const int h = blockIdx.y;
  const int b = blockIdx.z;
  const int m0 = it * 16;
  const int bh = b * kH + h;
  const int ncols = (it + 1) * 16;  // causal: only tiles j <= it

  // preload Q fragments for this wave's 16 rows (e offset 64 = q chunk)
  const size_t qbase = ((size_t)bh * kN + m0 + l16) * kE + kHD;
  v2f qa[16];
#pragma unroll
  for (int kk = 0; kk < 16; ++kk)
    qa[kk] = *(const v2f*)&kqv[qbase + kk * 4 + 2 * half];

  // ---- S = Q K^T / sqrt(hd), causal-masked, into LDS ----
  for (int j = w; j <= it; j += 4) {
    v8f acc = {};
    const size_t kbase = ((size_t)bh * kN + j * 16 + l16) * kE;  // k at e=0
#pragma unroll
    for (int kk = 0; kk < 16; ++kk) {
      v2f kb = *(const v2f*)&kqv[kbase + kk * 4 + 2 * half];
      acc = wmma_f32(qa[kk], kb, acc);
    }
#pragma unroll
    for (int v = 0; v < 8; ++v) {
      int row = v + 8 * half;
      int colg = j * 16 + l16;
      float sv = acc[v] * 0.125f;  // 1/sqrt(64)
      if (colg > m0 + row) sv = -INFINITY;
      s[row * SS + j * 16 + l16] = sv;
    }
  }
  __syncthreads();

  // ---- softmax stats: 8 threads per row ----
  {
    const int r = tid >> 3, g = tid & 7;
    float m = -INFINITY;
    for (int c = g; c < ncols; c += 8) m = fmaxf(m, s[r * SS + c]);
    red[r * 8 + g] = m;
    __syncthreads();
    if (g == 0) {
      float mm = red[r * 8];
      for (int i = 1; i < 8; ++i) mm = fmaxf(mm, red[r * 8 + i]);
      rowmax[r] = mm;
    }
    __syncthreads();
    float mm = rowmax[r];
    float ss = 0.f;
    for (int c = g; c < ncols; c += 8) ss += __expf(s[r * SS + c] - mm);
    red[r * 8 + g] = ss;
    __syncthreads();
    if (g == 0) {
      float t = 0.f;
      for (int i = 0; i < 8; ++i) t += red[r * 8 + i];
      rowsum[r] = t;
    }
    __syncthreads();
  }

  // ---- normalize, write A (and zero-fill the masked tail) ----
  float* Ab = Aout + ((size_t)bh * kN + m0) * kN;
  for (int row = 0; row < 16; ++row) {
    float mm = rowmax[row];
    float inv = 1.0f / rowsum[row];
    for (int col = tid; col < ncols; col += 128) {
      float p = __expf(s[row * SS + col] - mm) * inv;
      s[row * SS + col] = p;
      Ab[(size_t)row * kN + col] = p;
    }
    for (int col = ncols + tid; col < kN; col += 128)
      Ab[(size_t)row * kN + col] = 0.0f;
  }
  __syncthreads();

  // ---- O = A @ V (v chunk at e=128), per-wave partial over its tiles ----
  v8f oacc[4] = {};
  for (int j = w; j <= it; j += 4) {
#pragma unroll
    for (int kk = 0; kk < 4; ++kk) {
      v2f a = *(const v2f*)&s[l16 * SS + j * 16 + kk * 4 + 2 * half];
      int krow = j * 16 + kk * 4 + 2 * half;
      const float* vb = kqv + ((size_t)bh * kN + krow) * kE + 2 * kHD;
#pragma unroll
      for (int vt = 0; vt < 4; ++vt) {
        v2f bf;
        bf[0] = vb[vt * 16 + l16];
        bf[1] = vb[(size_t)kE + vt * 16 + l16];  // next V row
        oacc[vt] = wmma_f32(a, bf, oacc[vt]);
      }
    }
  }
#pragma unroll
  for (int vt = 0; vt < 4; ++vt)
#pragma unroll
    for (int v = 0; v < 8; ++v)
      ored[(w * 16 + v + 8 * half) * 64 + vt * 16 + l16] = oacc[vt][v];
  __syncthreads();

  // reduce 4 wave partials, write concat-heads sa[b, n, h*64 + dd]
  for (int i = tid; i < 16 * 64; i += 128) {
    float o = ored[i] + ored[1024 + i] + ored[2048 + i] + ored[3072 + i];
    int row = i >> 6, dd = i & 63;
    sa[((size_t)b * kN + m0 + row) * kD + h * kHD + dd] = o;
  }
}

// ---------------------------------------------------------------------------
// Kernel 3: out = sa @ Wproj^T + bproj.  [4096 x 1024] x [1024 x 1024]
// Block: 128 threads (4 waves), tile 64 x 64, double-buffered async staging.
// ---------------------------------------------------------------------------
__global__ __launch_bounds__(128) void proj_gemm_kernel(
    const float* __restrict__ sa, const float* __restrict__ Wproj,
    const float* __restrict__ bproj, float* __restrict__ out) {
  constexpr int TS = 68;
  __shared__ float xs[2][64 * TS];
  __shared__ float wt[2][64 * TS];

  const int tid = threadIdx.x;
  const int w = tid >> 5;
  const int lane = tid & 31;
  const int half = lane >> 4;
  const int l16 = lane & 15;

  const int r0 = blockIdx.x * 64;  // rows over flattened B*N
  const int c0 = blockIdx.y * 64;  // output feature cols

  auto stage = [&](int bufi, int kt) {
    for (int i = tid; i < 64 * 16; i += 128) {
      int r = i >> 4, c = (i & 15) * 4;
      lds_copy16(&sa[(size_t)(r0 + r) * kD + kt + c], &xs[bufi][r * TS + c]);
    }
    for (int i = tid; i < 64 * 16; i += 128) {
      int r = i >> 4, c = (i & 15) * 4;
      lds_copy16(&Wproj[(size_t)(c0 + r) * kD + kt + c], &wt[bufi][r * TS + c]);
    }
  };

  v8f acc[4] = {};

  stage(0, 0);
  async_stage_wait();
  __syncthreads();

  int buf = 0;
  for (int kt = 0; kt < kD; kt += 64) {
    if (kt + 64 < kD) stage(buf ^ 1, kt + 64);

#pragma unroll
    for (int kk = 0; kk < 64; kk += 4) {
      v2f a = *(const v2f*)&xs[buf][(w * 16 + l16) * TS + kk + 2 * half];
#pragma unroll
      for (int ct = 0; ct < 4; ++ct) {
        v2f bf = *(const v2f*)&wt[buf][(ct * 16 + l16) * TS + kk + 2 * half];
        acc[ct] = wmma_f32(a, bf, acc[ct]);
      }
    }
    async_stage_wait();
    __syncthreads();
    buf ^= 1;
  }

#pragma unroll
  for (int ct = 0; ct < 4; ++ct) {
    int col = c0 + ct * 16 + l16;
    float bias = bproj[col];
#pragma unroll
    for (int v = 0; v < 8; ++v) {
      int row = r0 + w * 16 + v + 8 * half;
      out[(size_t)row * kD + col] = acc[ct][v] + bias;
    }
  }
}

// ---------------------------------------------------------------------------
extern "C" void kernel_launch(void* const* d_in, const int* in_sizes, int n_in,
                              void* d_out, int out_size, void* d_ws,
                              size_t ws_size, hipStream_t stream) {
  const float* x = (const float*)d_in[0];
  const float* Wkqv = (const float*)d_in[1];
  const float* bkqv = (const float*)d_in[2];
  const float* Wproj = (const float*)d_in[3];
  const float* bproj = (const float*)d_in[4];

  float* out = (float*)d_out;             // [B,N,D]
  float* A = out + (size_t)kB * kN * kD;  // [B,H,N,N]

  float* kqvws = (float*)d_ws;                      // [B,H,N,192]  ~50.3 MB
  float* saws = kqvws + (size_t)kB * kH * kN * kE;  // [B,N,D]      ~16.8 MB

  kqv_gemm_kernel<<<dim3(kN / 64, kH, kB), 128, 0, stream>>>(x, Wkqv, bkqv,
                                                             kqvws);
  attn_kernel<<<dim3(kN / 16, kH, kB), 128, 0, stream>>>(kqvws, A, saws);
  proj_gemm_kernel<<<dim3(kB * kN / 64, kD / 64), 128, 0, stream>>>(
      saws, Wproj, bproj, out);
}